// GIN_76897094468094
// MI455X (gfx1250) — compile-verified
//
#include <hip/hip_runtime.h>
#include <math.h>

#define N_NODES  100000
#define N_EDGES  1600000
#define N_GRAPHS 2000
#define IN_DIM   128
#define HID      64

typedef __attribute__((ext_vector_type(2))) float v2f;
typedef __attribute__((ext_vector_type(8))) float v8f;

__device__ __forceinline__ float gelu_exact(float x) {
    return 0.5f * x * (1.0f + erff(x * 0.70710678118654752440f));
}

// order-preserving encode of float into uint for atomicMax (handles negatives)
__device__ __forceinline__ unsigned fenc(float x) {
    unsigned u = __float_as_uint(x);
    return (u & 0x80000000u) ? ~u : (u | 0x80000000u);
}
__device__ __forceinline__ float fdec(unsigned u) {
    return __uint_as_float((u & 0x80000000u) ? (u & 0x7FFFFFFFu) : ~u);
}

__global__ __launch_bounds__(256) void k_fill_u32(unsigned* p, int n, unsigned v) {
    int i = blockIdx.x * 256 + threadIdx.x;
    if (i < n) p[i] = v;
}

__global__ __launch_bounds__(256)
void k_input_bn(const float* __restrict__ x, const float* __restrict__ g,
                const float* __restrict__ b, const float* __restrict__ m,
                const float* __restrict__ v, float* __restrict__ out) {
    int i = blockIdx.x * 256 + threadIdx.x;
    if (i >= N_NODES * IN_DIM) return;
    int c = i & 127;
    out[i] = (x[i] - m[c]) * rsqrtf(v[c] + 1e-5f) * g[c] + b[c];
}

// one wave handles 1 edge (dim=128) or 2 edges (dim=64); float4 per lane
__global__ __launch_bounds__(256)
void k_scatter_add(const float* __restrict__ X, const int* __restrict__ src,
                   const int* __restrict__ dst, float* __restrict__ AGG, int shift) {
    int t = blockIdx.x * 256 + threadIdx.x;
    if (t >= (N_EDGES << shift)) return;
    int e  = t >> shift;
    int c4 = t & ((1 << shift) - 1);
    int dim = 4 << shift;
    int s = src[e], d = dst[e];
    const float4 xv = *(const float4*)(X + (size_t)s * dim + c4 * 4);
    float* ap = AGG + (size_t)d * dim + c4 * 4;
    atomicAdd(ap + 0, xv.x);
    atomicAdd(ap + 1, xv.y);
    atomicAdd(ap + 2, xv.z);
    atomicAdd(ap + 3, xv.w);
}

// Fused GIN layer MLP over a 16-node tile:
//   h=(1+eps)x+agg -> lin1(WMMA f32) -> LN(128) -> gelu -> lin2(WMMA f32)
//   -> LN(64) -> gelu -> BN -> XOUT, JSUM += softmax(jump)[l] * XOUT
__global__ __launch_bounds__(256)
void k_gin_mlp(const float* __restrict__ X, const float* __restrict__ AGG,
               const float* __restrict__ epsp,
               const float* __restrict__ W1, const float* __restrict__ B1,
               const float* __restrict__ G1, const float* __restrict__ Be1,
               const float* __restrict__ W2, const float* __restrict__ B2,
               const float* __restrict__ G2, const float* __restrict__ Be2,
               const float* __restrict__ BNG, const float* __restrict__ BNB,
               const float* __restrict__ BNM, const float* __restrict__ BNV,
               const float* __restrict__ jumpv, int layer, int K,
               float* __restrict__ XOUT, float* __restrict__ JSUM) {
    __shared__ float Atile[16 * 128];
    __shared__ float Htile[16 * 128];
    __shared__ float H2[16 * 64];
    int tid = threadIdx.x;
    int m0  = blockIdx.x * 16;
    float epsv = 1.0f + epsp[0];

    for (int idx = tid; idx < 16 * K; idx += 256) {
        int r = idx / K, k = idx - r * K;
        size_t gi = (size_t)(m0 + r) * K + k;
        Atile[idx] = epsv * X[gi] + AGG[gi];
    }
    __syncthreads();

    int wave = tid >> 5, lane = tid & 31, h = lane >> 4, i = lane & 15;

    { // GEMM1: [16xK] @ [Kx128], each wave owns a 16-col slab
        int n0 = wave * 16;
        v8f c = {0.f, 0.f, 0.f, 0.f, 0.f, 0.f, 0.f, 0.f};
        for (int k = 0; k < K; k += 4) {
            int ka = k + 2 * h;
            v2f a, b;
            a.x = Atile[i * K + ka];
            a.y = Atile[i * K + ka + 1];
            b.x = W1[ka * 128 + n0 + i];
            b.y = W1[(ka + 1) * 128 + n0 + i];
            c = __builtin_amdgcn_wmma_f32_16x16x4_f32(false, a, false, b, (short)0, c,
                                                      false, false);
        }
        for (int r = 0; r < 8; ++r)
            Htile[(r + 8 * h) * 128 + n0 + i] = c[r] + B1[n0 + i];
    }
    __syncthreads();

    { // LN(128) + gelu: 16 threads (half-wave) per row, 8 cols each
        int row = tid >> 4, sub = tid & 15;
        float s = 0.f, ss = 0.f;
        for (int j = 0; j < 8; ++j) {
            float v = Htile[row * 128 + sub * 8 + j];
            s += v; ss += v * v;
        }
        for (int off = 8; off >= 1; off >>= 1) {
            s  += __shfl_xor(s, off, 16);
            ss += __shfl_xor(ss, off, 16);
        }
        float mu  = s * (1.f / 128.f);
        float inv = rsqrtf(ss * (1.f / 128.f) - mu * mu + 1e-5f);
        for (int j = 0; j < 8; ++j) {
            int cc = sub * 8 + j;
            float v = (Htile[row * 128 + cc] - mu) * inv * G1[cc] + Be1[cc];
            Htile[row * 128 + cc] = gelu_exact(v);
        }
    }
    __syncthreads();

    if (wave < 4) { // GEMM2: [16x128] @ [128x64]
        int n0 = wave * 16;
        v8f c = {0.f, 0.f, 0.f, 0.f, 0.f, 0.f, 0.f, 0.f};
        for (int k = 0; k < 128; k += 4) {
            int ka = k + 2 * h;
            v2f a, b;
            a.x = Htile[i * 128 + ka];
            a.y = Htile[i * 128 + ka + 1];
            b.x = W2[ka * 64 + n0 + i];
            b.y = W2[(ka + 1) * 64 + n0 + i];
            c = __builtin_amdgcn_wmma_f32_16x16x4_f32(false, a, false, b, (short)0, c,
                                                      false, false);
        }
        for (int r = 0; r < 8; ++r)
            H2[(r + 8 * h) * 64 + n0 + i] = c[r] + B2[n0 + i];
    }
    __syncthreads();

    { // LN(64) + gelu + BN + write + jump accumulate
        int row = tid >> 4, sub = tid & 15;
        float s = 0.f, ss = 0.f;
        for (int j = 0; j < 4; ++j) {
            float v = H2[row * 64 + sub * 4 + j];
            s += v; ss += v * v;
        }
        for (int off = 8; off >= 1; off >>= 1) {
            s  += __shfl_xor(s, off, 16);
            ss += __shfl_xor(ss, off, 16);
        }
        float mu  = s * (1.f / 64.f);
        float inv = rsqrtf(ss * (1.f / 64.f) - mu * mu + 1e-5f);
        float j0 = jumpv[0], j1 = jumpv[1], j2 = jumpv[2];
        float mj = fmaxf(j0, fmaxf(j1, j2));
        float e0 = expf(j0 - mj), e1 = expf(j1 - mj), e2 = expf(j2 - mj);
        float jw = ((layer == 0) ? e0 : (layer == 1) ? e1 : e2) / (e0 + e1 + e2);
        for (int j = 0; j < 4; ++j) {
            int cc = sub * 4 + j;
            float v = (H2[row * 64 + cc] - mu) * inv * G2[cc] + Be2[cc];
            v = gelu_exact(v);
            v = (v - BNM[cc]) * rsqrtf(BNV[cc] + 1e-5f) * BNG[cc] + BNB[cc];
            size_t oi = (size_t)(m0 + row) * 64 + cc;
            XOUT[oi] = v;
            float prev = (layer == 0) ? 0.f : JSUM[oi];
            JSUM[oi] = prev + jw * v;
        }
    }
}

__global__ __launch_bounds__(256)
void k_pool_scatter(const float* __restrict__ JS, const int* __restrict__ batch,
                    float* __restrict__ SSUM, float* __restrict__ CNT,
                    unsigned* __restrict__ SMAXE) {
    int t = blockIdx.x * 256 + threadIdx.x;
    if (t >= N_NODES * 64) return;
    int n = t >> 6, c = t & 63;
    int g = batch[n];
    float v = JS[t];
    atomicAdd(&SSUM[g * 64 + c], v);
    atomicMax(&SMAXE[g * 64 + c], fenc(v));
    if (c == 0) atomicAdd(&CNT[g], 1.0f);
}

__global__ __launch_bounds__(256)
void k_pool_fin(const float* __restrict__ SSUM, const float* __restrict__ CNT,
                const unsigned* __restrict__ SMAXE, const float* __restrict__ pw,
                float* __restrict__ POOL) {
    int t = blockIdx.x * 256 + threadIdx.x;
    if (t >= N_GRAPHS * 64) return;
    int g = t >> 6;
    float p0 = pw[0], p1 = pw[1], p2 = pw[2];
    float mp = fmaxf(p0, fmaxf(p1, p2));
    float e0 = expf(p0 - mp), e1 = expf(p1 - mp), e2 = expf(p2 - mp);
    float inv = 1.f / (e0 + e1 + e2);
    float cnt   = CNT[g];
    float ssum  = SSUM[t];
    float smean = ssum / fmaxf(cnt, 1.0f);
    float smax  = (cnt > 0.f) ? fdec(SMAXE[t]) : 0.f;
    POOL[t] = ssum * (e0 * inv) + smean * (e1 * inv) + smax * (e2 * inv);
}

// head: out = gelu(LN(P@fc1w+fc1b)) + P ; out @ fc2w + fc2b  (16 graphs/block)
__global__ __launch_bounds__(256)
void k_head(const float* __restrict__ POOL, const float* __restrict__ FC1W,
            const float* __restrict__ FC1B, const float* __restrict__ LNG,
            const float* __restrict__ LNB, const float* __restrict__ FC2W,
            const float* __restrict__ FC2B, float* __restrict__ OUT) {
    __shared__ float P[16 * 64];
    __shared__ float T[16 * 64];
    int tid = threadIdx.x;
    int g0  = blockIdx.x * 16;
    for (int idx = tid; idx < 16 * 64; idx += 256) P[idx] = POOL[(size_t)g0 * 64 + idx];
    __syncthreads();
    int wave = tid >> 5, lane = tid & 31, h = lane >> 4, i = lane & 15;
    if (wave < 4) {
        int n0 = wave * 16;
        v8f c = {0.f, 0.f, 0.f, 0.f, 0.f, 0.f, 0.f, 0.f};
        for (int k = 0; k < 64; k += 4) {
            int ka = k + 2 * h;
            v2f a, b;
            a.x = P[i * 64 + ka];  a.y = P[i * 64 + ka + 1];
            b.x = FC1W[ka * 64 + n0 + i];  b.y = FC1W[(ka + 1) * 64 + n0 + i];
            c = __builtin_amdgcn_wmma_f32_16x16x4_f32(false, a, false, b, (short)0, c,
                                                      false, false);
        }
        for (int r = 0; r < 8; ++r) T[(r + 8 * h) * 64 + n0 + i] = c[r] + FC1B[n0 + i];
    }
    __syncthreads();
    {
        int row = tid >> 4, sub = tid & 15;
        float s = 0.f, ss = 0.f;
        for (int j = 0; j < 4; ++j) {
            float v = T[row * 64 + sub * 4 + j];
            s += v; ss += v * v;
        }
        for (int off = 8; off >= 1; off >>= 1) {
            s  += __shfl_xor(s, off, 16);
            ss += __shfl_xor(ss, off, 16);
        }
        float mu  = s * (1.f / 64.f);
        float inv = rsqrtf(ss * (1.f / 64.f) - mu * mu + 1e-5f);
        for (int j = 0; j < 4; ++j) {
            int cc = sub * 4 + j;
            float v = (T[row * 64 + cc] - mu) * inv * LNG[cc] + LNB[cc];
            P[row * 64 + cc] = gelu_exact(v) + P[row * 64 + cc];
        }
    }
    __syncthreads();
    if (wave < 2) {
        int n0 = wave * 16;
        v8f c = {0.f, 0.f, 0.f, 0.f, 0.f, 0.f, 0.f, 0.f};
        for (int k = 0; k < 64; k += 4) {
            int ka = k + 2 * h;
            v2f a, b;
            a.x = P[i * 64 + ka];  a.y = P[i * 64 + ka + 1];
            b.x = FC2W[ka * 32 + n0 + i];  b.y = FC2W[(ka + 1) * 32 + n0 + i];
            c = __builtin_amdgcn_wmma_f32_16x16x4_f32(false, a, false, b, (short)0, c,
                                                      false, false);
        }
        for (int r = 0; r < 8; ++r)
            OUT[(size_t)(g0 + r + 8 * h) * 32 + n0 + i] = c[r] + FC2B[n0 + i];
    }
}

extern "C" void kernel_launch(void* const* d_in, const int* in_sizes, int n_in,
                              void* d_out, int out_size, void* d_ws, size_t ws_size,
                              hipStream_t stream) {
    const float* x   = (const float*)d_in[0];
    const int* ei    = (const int*)d_in[1];
    const int* src   = ei;            // edge_index[0]
    const int* dst   = ei + N_EDGES;  // edge_index[1]
    const int* batch = (const int*)d_in[2];
    auto F = [&](int i) { return (const float*)d_in[i]; };
    // params flattened in insertion order:
    // 3..6: input_bn g,b,m,v
    // per layer (base 7/20/33): eps, lin1.w, lin1.b, ln1.g, ln1.b,
    //                           lin2.w, lin2.b, ln2.g, ln2.b, bn.g, bn.b, bn.m, bn.v
    // 46 jump, 47 pool_w, 48/49 fc1 w/b, 50/51 ln_out g/b, 52/53 fc2 w/b

    float* ws = (float*)d_ws;
    float*    AGG   = ws;                    // 12.8M f
    float*    BUFA  = ws + 12800000;         // 12.8M f
    float*    BUFB  = ws + 25600000;         // 6.4M f
    float*    JSUM  = ws + 32000000;         // 6.4M f
    float*    SSUM  = ws + 38400000;         // 128k f
    float*    CNT   = ws + 38528000;         // 2k f
    unsigned* SMAXE = (unsigned*)(ws + 38530000); // 128k u32
    float*    POOL  = ws + 38658000;         // 128k f

    dim3 B(256);
    k_input_bn<<<(N_NODES * IN_DIM + 255) / 256, B, 0, stream>>>(x, F(3), F(4), F(5),
                                                                 F(6), BUFA);

    struct { const float* X; float* XO; int K; int base; } Ls[3] = {
        {BUFA, BUFB, 128, 7}, {BUFB, BUFA, 64, 20}, {BUFA, BUFB, 64, 33}};
    for (int l = 0; l < 3; ++l) {
        int K = Ls[l].K, b = Ls[l].base;
        int aggN = N_NODES * K;
        k_fill_u32<<<(aggN + 255) / 256, B, 0, stream>>>((unsigned*)AGG, aggN, 0u);
        int shift = (K == 128) ? 5 : 4;
        int tot = N_EDGES << shift;
        k_scatter_add<<<(tot + 255) / 256, B, 0, stream>>>(Ls[l].X, src, dst, AGG, shift);
        k_gin_mlp<<<N_NODES / 16, B, 0, stream>>>(
            Ls[l].X, AGG, F(b + 0), F(b + 1), F(b + 2), F(b + 3), F(b + 4), F(b + 5),
            F(b + 6), F(b + 7), F(b + 8), F(b + 9), F(b + 10), F(b + 11), F(b + 12),
            F(46), l, K, Ls[l].XO, JSUM);
    }

    k_fill_u32<<<(N_GRAPHS * 64 + 255) / 256, B, 0, stream>>>((unsigned*)SSUM,
                                                              N_GRAPHS * 64, 0u);
    k_fill_u32<<<(N_GRAPHS + 255) / 256, B, 0, stream>>>((unsigned*)CNT, N_GRAPHS, 0u);
    k_fill_u32<<<(N_GRAPHS * 64 + 255) / 256, B, 0, stream>>>(SMAXE, N_GRAPHS * 64,
                                                              0x007FFFFFu); // enc(-inf)
    k_pool_scatter<<<(N_NODES * 64 + 255) / 256, B, 0, stream>>>(JSUM, batch, SSUM, CNT,
                                                                 SMAXE);
    k_pool_fin<<<(N_GRAPHS * 64 + 255) / 256, B, 0, stream>>>(SSUM, CNT, SMAXE, F(47),
                                                              POOL);
    k_head<<<N_GRAPHS / 16, B, 0, stream>>>(POOL, F(48), F(49), F(50), F(51), F(52),
                                            F(53), (float*)d_out);
}